// PeakResponseLocalizer_79757542687318
// MI455X (gfx1250) — compile-verified
//
#include <hip/hip_runtime.h>

typedef __attribute__((ext_vector_type(2))) float v2f;
typedef __attribute__((ext_vector_type(4))) float v4f;
typedef __attribute__((ext_vector_type(8))) float v8f;
typedef int b128i __attribute__((vector_size(16)));   // builtin's b128 payload type

#define CDIM   1024
#define HWDIM  4096   // 64*64
#define NCLS   19
#define MPAD   32
#define KC     64     // C-chunk staged in LDS
#define TILE_N 128    // 8 waves * 16 columns
#define FPAD   136    // row stride: 2*136 % 64 == 16 -> conflict-free half-wave split

#if defined(__has_builtin)
#  if __has_builtin(__builtin_amdgcn_global_load_async_to_lds_b128) && \
      __has_builtin(__builtin_amdgcn_s_wait_asynccnt)
#    define USE_ASYNC_LDS 1
#  endif
#endif
#ifndef USE_ASYNC_LDS
#  define USE_ASYNC_LDS 0
#endif

// Stage one [KC x TILE_N] f32 FM chunk into LDS: 2048 x b128, 8 per thread.
// Whole-wave writes one contiguous 512B row segment -> conflict-free.
__device__ __forceinline__ void stage_fm_chunk(const float* __restrict__ fmb,
                                               int c0, int n0,
                                               float (*dst)[FPAD], int tid) {
#pragma unroll
  for (int i = 0; i < (KC * TILE_N) / (4 * 256); ++i) {   // 8 iters
    int idx = tid + i * 256;        // b128 unit id, 0..2047
    int r   = idx >> 5;             // 0..63  (c row)
    int c4  = (idx & 31) << 2;      // 0..124 (hw col, x4 floats)
    const float* g = fmb + (size_t)(c0 + r) * HWDIM + n0 + c4;
    float* l = &dst[r][c4];
#if USE_ASYNC_LDS
    __builtin_amdgcn_global_load_async_to_lds_b128(
        (__attribute__((address_space(1))) b128i*)g,
        (__attribute__((address_space(3))) b128i*)l, 0, 0);
#else
    *(v4f*)l = *(const v4f*)g;      // global_load_b128 + ds_store_b128
#endif
  }
}

__device__ __forceinline__ void wait_stage_done() {
#if USE_ASYNC_LDS
  __builtin_amdgcn_s_wait_asynccnt(0);
#endif
  __syncthreads();
}

// ---------------------------------------------------------------------------
// Kernel 1: cams[b,k,hw] = sum_c FM[b,c,hw] * W[k,c] + bias[k]
// WG = (batch, 128 hw cols); 8 waves each own a 16-col N tile and both 16-row
// M tiles (classes padded 19->32, dead rows zero) sharing one B fragment, so
// FM streams from HBM exactly once, via double-buffered async LDS staging.
// ---------------------------------------------------------------------------
__global__ __launch_bounds__(256) void cams_gemm_wmma(
    const float* __restrict__ fm,     // [8][1024][4096]
    const float* __restrict__ wgt,    // [19][1024]
    const float* __restrict__ bias,   // [19]
    float* __restrict__ cams)         // [8][19][4096]
{
  __shared__ float wlds[MPAD][KC + 1];      // weights chunk, padded
  __shared__ float fbuf[2][KC][FPAD];       // double-buffered FM chunk

  const int b    = blockIdx.y;
  const int n0   = blockIdx.x * TILE_N;
  const int tid  = threadIdx.x;
  const int lane = tid & 31;
  const int wave = tid >> 5;                // 0..7 -> N-tile
  const int colN = lane & 15;               // N col / M row within 16-tile
  const int kkb  = (lane >> 4) << 1;        // 0 (lanes 0-15) or 2 (lanes 16-31)
  const int hwl  = wave * 16 + colN;        // local hw column 0..127
  const int hw   = n0 + hwl;

  const float* fmb = fm + (size_t)b * CDIM * HWDIM;

  v8f acc0 = {};                            // classes 0..15
  v8f acc1 = {};                            // classes 16..31 (19..31 dead)

  int buf = 0;
  stage_fm_chunk(fmb, 0, n0, fbuf[0], tid); // preload chunk 0

  for (int c0 = 0; c0 < CDIM; c0 += KC) {
    // stage W chunk [32][64] (rows >= 19 zeroed)
#pragma unroll
    for (int i = 0; i < (MPAD * KC) / 256; ++i) {
      int idx = tid + i * 256;
      int r   = idx >> 6;
      int cc  = idx & 63;
      wlds[r][cc] = (r < NCLS) ? wgt[r * CDIM + c0 + cc] : 0.0f;
    }
    wait_stage_done();                      // fbuf[buf] + wlds ready everywhere

    if (c0 + KC < CDIM)                     // kick DMA for next chunk
      stage_fm_chunk(fmb, c0 + KC, n0, fbuf[buf ^ 1], tid);

#pragma unroll 4
    for (int kk = 0; kk < KC; kk += 4) {
      v2f a0, a1, bf2;
      // A frag (ISA: lanes 0-15 hold K0/K1, lanes 16-31 K2/K3)
      a0.x = wlds[colN][kk + kkb];
      a0.y = wlds[colN][kk + kkb + 1];
      a1.x = wlds[16 + colN][kk + kkb];
      a1.y = wlds[16 + colN][kk + kkb + 1];
      // B frag from staged LDS tile (mirror layout on columns)
      bf2.x = fbuf[buf][kk + kkb][hwl];
      bf2.y = fbuf[buf][kk + kkb + 1][hwl];
      acc0 = __builtin_amdgcn_wmma_f32_16x16x4_f32(
          false, a0, false, bf2, (short)0, acc0, false, false);
      acc1 = __builtin_amdgcn_wmma_f32_16x16x4_f32(
          false, a1, false, bf2, (short)0, acc1, false, false);
    }
    __syncthreads();                        // protect wlds/fbuf reuse
    buf ^= 1;
  }

  // store C/D: VGPR i holds M = i + 8*(lane>=16), N = lane&15
  const int mloc = (lane >> 4) * 8;
#pragma unroll
  for (int i = 0; i < 8; ++i) {
    int cls = mloc + i;
    if (cls < NCLS)
      cams[((size_t)(b * NCLS + cls)) * HWDIM + hw] = acc0[i] + bias[cls];
    int cls1 = 16 + mloc + i;
    if (cls1 < NCLS)
      cams[((size_t)(b * NCLS + cls1)) * HWDIM + hw] = acc1[i] + bias[cls1];
  }
}

// ---------------------------------------------------------------------------
// Kernel 2: per (b,k): lower median (rank 2047 of 4096) via 32-pass radix
// select on monotone float keys, then 3x3 peak test and num/den reduction.
// ---------------------------------------------------------------------------
__global__ __launch_bounds__(256) void peak_logits_kernel(
    const float* __restrict__ cams,   // [8*19][4096]
    float* __restrict__ out)          // [8*19]
{
  __shared__ float sv[HWDIM];
  __shared__ int   cnt;
  __shared__ float rnum, rden;

  const int bk = blockIdx.x;
  const int t  = threadIdx.x;
  const float* src = cams + (size_t)bk * HWDIM;

  for (int i = t; i < HWDIM; i += 256) sv[i] = src[i];
  if (t == 0) { rnum = 0.0f; rden = 0.0f; }
  __syncthreads();

  // radix select: element of sorted rank 2047 (lower median)
  unsigned prefix = 0u;
  int r = (HWDIM - 1) >> 1;
  for (int bit = 31; bit >= 0; --bit) {
    if (t == 0) cnt = 0;
    __syncthreads();
    int local = 0;
    for (int i = t; i < HWDIM; i += 256) {
      unsigned u    = __float_as_uint(sv[i]);
      unsigned key  = (u & 0x80000000u) ? ~u : (u | 0x80000000u);
      unsigned diff = key ^ prefix;
      if (((diff >> bit) >> 1) == 0u && ((key >> bit) & 1u) == 0u) ++local;
    }
    atomicAdd(&cnt, local);
    __syncthreads();
    int c0 = cnt;
    if (r >= c0) { prefix |= (1u << bit); r -= c0; }
    __syncthreads();
  }
  unsigned mu = (prefix & 0x80000000u) ? (prefix ^ 0x80000000u) : ~prefix;
  const float med = __uint_as_float(mu);

  // 3x3 local-max peak test with implicit -inf padding
  float num = 0.0f, den = 0.0f;
  for (int i = t; i < HWDIM; i += 256) {
    int h = i >> 6, w = i & 63;
    float m = sv[i];
    float lmax = m;
#pragma unroll
    for (int dh = -1; dh <= 1; ++dh) {
      int hh = h + dh;
      if (hh < 0 || hh > 63) continue;
#pragma unroll
      for (int dw = -1; dw <= 1; ++dw) {
        int ww = w + dw;
        if (ww < 0 || ww > 63) continue;
        float v = sv[(hh << 6) | ww];
        lmax = fmaxf(lmax, v);
      }
    }
    if (m == lmax && m >= med) { num += m; den += 1.0f; }
  }
  atomicAdd(&rnum, num);
  atomicAdd(&rden, den);
  __syncthreads();
  if (t == 0) out[bk] = rnum / rden;
}

// ---------------------------------------------------------------------------
extern "C" void kernel_launch(void* const* d_in, const int* in_sizes, int n_in,
                              void* d_out, int out_size, void* d_ws, size_t ws_size,
                              hipStream_t stream) {
  const float* fm   = (const float*)d_in[0];   // [8,1024,64,64]
  const float* wgt  = (const float*)d_in[1];   // [19,1024,1,1]
  const float* bias = (const float*)d_in[2];   // [19]
  float* out  = (float*)d_out;                 // [8,19]
  float* cams = (float*)d_ws;                  // [8,19,4096] scratch (2.49 MB)

  dim3 g1(HWDIM / TILE_N, 8);                  // 32 x 8 = 256 WGs
  cams_gemm_wmma<<<g1, 256, 0, stream>>>(fm, wgt, bias, cams);

  peak_logits_kernel<<<dim3(8 * NCLS), 256, 0, stream>>>(cams, out);
}